// QGIN_22239340659447
// MI455X (gfx1250) — compile-verified
//
#include <hip/hip_runtime.h>
#include <hip/hip_bf16.h>
#include <math.h>

#define N_NODES 50000
#define N_EDGES 800000
#define N_GRAPHS 1000
#define F_IN 128
#define HID 64
#define OUTF 16
#define BN_EPS 1e-5f

typedef __attribute__((ext_vector_type(2))) float v2f;
typedef __attribute__((ext_vector_type(8))) float v8f;

// ---- order-preserving float<->uint map for atomicMax on floats ----
__device__ __forceinline__ unsigned f2ord(float f) {
    unsigned u = __float_as_uint(f);
    return (u & 0x80000000u) ? ~u : (u | 0x80000000u);
}
__device__ __forceinline__ float ord2f(unsigned e) {
    unsigned u = (e & 0x80000000u) ? (e & 0x7FFFFFFFu) : ~e;
    return __uint_as_float(u);
}

// ---- copy src->dst and zero the BN stats buffer ----
__global__ void k_prep(float* __restrict__ dst, const float* __restrict__ src,
                       int count, float* __restrict__ stats) {
    int i = blockIdx.x * blockDim.x + threadIdx.x;
    if (i < count) dst[i] = src[i];
    if (i < 2 * HID) stats[i] = 0.0f;
}

// ---- edge scatter: y[dst] += x[src]; one wave per edge ----
template <int F>
__global__ void k_edge(float* __restrict__ y, const float* __restrict__ x,
                       const int* __restrict__ ei) {
    int e = blockIdx.x * (blockDim.x >> 5) + (threadIdx.x >> 5);
    if (e >= N_EDGES) return;
    int lane = threadIdx.x & 31;
    int s = ei[e];
    int d = ei[N_EDGES + e];
    const float* xs = x + (long)s * F;
    float* yd = y + (long)d * F;
    for (int f = lane; f < F; f += 32) atomicAdd(&yd[f], xs[f]);
}

// ---- fused MLP: relu(relu(X@W1+b1)@W2+b2) with fp32 WMMA, + BN stats ----
// One 16-row tile per wave; 50000 % 16 == 0, so active waves have full EXEC.
template <int K1>
__global__ void __launch_bounds__(256) k_mlp(
    const float* __restrict__ X, const float* __restrict__ W1,
    const float* __restrict__ B1, const float* __restrict__ W2,
    const float* __restrict__ B2, float* __restrict__ H,
    float* __restrict__ stats) {
    __shared__ float lds[8 * 16 * HID];   // 32 KB: per-wave 16x64 h1 tile
    const int wv   = threadIdx.x >> 5;
    const int lane = threadIdx.x & 31;
    const int lhi  = lane >> 4;           // 0: K+0..1 / M 0..7 ; 1: K+2..3 / M 8..15
    const int llo  = lane & 15;
    const int tile = blockIdx.x * 8 + wv;
    const bool active = tile < (N_NODES / 16);
    float* myLds = lds + wv * 16 * HID;
    v8f zero = {};

    if (active) {
        const int rowBase = tile * 16;
        v8f acc[4];
        #pragma unroll
        for (int n = 0; n < 4; ++n) acc[n] = zero;
        const float* arow = X + (long)(rowBase + llo) * K1 + lhi * 2;
        for (int kk = 0; kk < K1; kk += 4) {
            v2f a;
            a.x = arow[kk];
            a.y = arow[kk + 1];
            #pragma unroll
            for (int n = 0; n < 4; ++n) {
                int col = n * 16 + llo;
                v2f b;
                b.x = W1[(kk + lhi * 2) * HID + col];
                b.y = W1[(kk + lhi * 2 + 1) * HID + col];
                acc[n] = __builtin_amdgcn_wmma_f32_16x16x4_f32(
                    false, a, false, b, (short)0, acc[n], false, false);
            }
        }
        // bias + relu -> LDS tile (row-major 16 x 64)
        #pragma unroll
        for (int n = 0; n < 4; ++n) {
            int col = n * 16 + llo;
            float bias = B1[col];
            #pragma unroll
            for (int v = 0; v < 8; ++v) {
                int m = v + lhi * 8;
                myLds[m * HID + col] = fmaxf(acc[n][v] + bias, 0.0f);
            }
        }
    }
    __syncthreads();
    if (active) {
        const int rowBase = tile * 16;
        v8f acc[4];
        #pragma unroll
        for (int n = 0; n < 4; ++n) acc[n] = zero;
        const float* arow = myLds + llo * HID + lhi * 2;
        for (int kk = 0; kk < HID; kk += 4) {
            v2f a;
            a.x = arow[kk];
            a.y = arow[kk + 1];
            #pragma unroll
            for (int n = 0; n < 4; ++n) {
                int col = n * 16 + llo;
                v2f b;
                b.x = W2[(kk + lhi * 2) * HID + col];
                b.y = W2[(kk + lhi * 2 + 1) * HID + col];
                acc[n] = __builtin_amdgcn_wmma_f32_16x16x4_f32(
                    false, a, false, b, (short)0, acc[n], false, false);
            }
        }
        // bias + relu, store h2, accumulate column sum / sumsq
        #pragma unroll
        for (int n = 0; n < 4; ++n) {
            int col = n * 16 + llo;
            float bias = B2[col];
            float s = 0.0f, ss = 0.0f;
            #pragma unroll
            for (int v = 0; v < 8; ++v) {
                int m = v + lhi * 8;
                float h = fmaxf(acc[n][v] + bias, 0.0f);
                H[(long)(rowBase + m) * HID + col] = h;
                s += h; ss += h * h;
            }
            s  += __shfl_xor(s, 16, 32);   // combine M 0..7 with M 8..15
            ss += __shfl_xor(ss, 16, 32);
            if (lhi == 0) {
                atomicAdd(&stats[col], s);
                atomicAdd(&stats[HID + col], ss);
            }
        }
    }
}

// ---- BN apply (training-mode batch stats, biased variance) ----
__global__ void k_bn(const float* __restrict__ H, const float* __restrict__ stats,
                     const float* __restrict__ gamma, const float* __restrict__ beta,
                     float* __restrict__ Y) {
    int i = blockIdx.x * blockDim.x + threadIdx.x;
    if (i >= N_NODES * HID) return;
    int c = i & (HID - 1);
    const float invN = 1.0f / (float)N_NODES;
    float mu = stats[c] * invN;
    float var = stats[HID + c] * invN - mu * mu;
    float scale = gamma[c] * rsqrtf(var + BN_EPS);
    Y[i] = (H[i] - mu) * scale + beta[c];
}

__global__ void k_pool_init(unsigned* __restrict__ p) {
    int i = blockIdx.x * blockDim.x + threadIdx.x;
    if (i < N_GRAPHS * HID) p[i] = f2ord(-__builtin_inff());
}

__global__ void k_pool(const float* __restrict__ X, const int* __restrict__ batch,
                       unsigned* __restrict__ p) {
    int i = blockIdx.x * blockDim.x + threadIdx.x;
    if (i >= N_NODES * HID) return;
    int node = i >> 6;
    int c = i & (HID - 1);
    int g = batch[node];
    atomicMax(&p[g * HID + c], f2ord(X[i]));
}

// ---- readout head: relu(pooled@lin1+b1)@lin2+b2 ; one 64-thread block/graph
__global__ void __launch_bounds__(64) k_head(
    const unsigned* __restrict__ p, const float* __restrict__ W1,
    const float* __restrict__ B1, const float* __restrict__ W2,
    const float* __restrict__ B2, float* __restrict__ out) {
    __shared__ float h0[HID];
    __shared__ float h1[HID];
    int g = blockIdx.x;
    int f = threadIdx.x;
    float v = ord2f(p[g * HID + f]);
    if (!isfinite(v)) v = 0.0f;   // empty segments -> 0
    h0[f] = v;
    __syncthreads();
    float acc = B1[f];
    for (int k = 0; k < HID; ++k) acc += h0[k] * W1[k * HID + f];
    h1[f] = fmaxf(acc, 0.0f);
    __syncthreads();
    if (f < OUTF) {
        float o = B2[f];
        for (int k = 0; k < HID; ++k) o += h1[k] * W2[k * OUTF + f];
        out[g * OUTF + f] = o;
    }
}

extern "C" void kernel_launch(void* const* d_in, const int* in_sizes, int n_in,
                              void* d_out, int out_size, void* d_ws, size_t ws_size,
                              hipStream_t stream) {
    const float* x      = (const float*)d_in[0];
    const int*   ei     = (const int*)d_in[1];
    const int*   batch  = (const int*)d_in[2];
    const float* l0_w1  = (const float*)d_in[3];
    const float* l0_b1  = (const float*)d_in[4];
    const float* l0_w2  = (const float*)d_in[5];
    const float* l0_b2  = (const float*)d_in[6];
    const float* l0_g   = (const float*)d_in[7];
    const float* l0_bt  = (const float*)d_in[8];
    const float* ws1    = (const float*)d_in[9];
    const float* bs1    = (const float*)d_in[10];
    const float* ws2    = (const float*)d_in[11];
    const float* bs2    = (const float*)d_in[12];
    const float* gammas = (const float*)d_in[13];
    const float* betas  = (const float*)d_in[14];
    const float* lin1_w = (const float*)d_in[15];
    const float* lin1_b = (const float*)d_in[16];
    const float* lin2_w = (const float*)d_in[17];
    const float* lin2_b = (const float*)d_in[18];
    float* out = (float*)d_out;

    // workspace carve-out (floats)
    float*    ws     = (float*)d_ws;
    float*    stats  = ws;                                        // 128
    float*    yAgg   = ws + 128;                                  // 50000*128
    float*    hBuf   = yAgg + (size_t)N_NODES * F_IN;             // 50000*64
    float*    xBuf   = hBuf + (size_t)N_NODES * HID;              // 50000*64
    unsigned* pooled = (unsigned*)(xBuf + (size_t)N_NODES * HID); // 1000*64

    const int TB = 256;
    auto cdiv = [](long a, long b) { return (int)((a + b - 1) / b); };

    // ---- layer 0 (F_IN -> HID) ----
    k_prep<<<cdiv((long)N_NODES * F_IN, TB), TB, 0, stream>>>(yAgg, x, N_NODES * F_IN, stats);
    k_edge<F_IN><<<cdiv(N_EDGES, 8), TB, 0, stream>>>(yAgg, x, ei);
    k_mlp<F_IN><<<cdiv(N_NODES / 16, 8), TB, 0, stream>>>(yAgg, l0_w1, l0_b1, l0_w2, l0_b2,
                                                          hBuf, stats);
    k_bn<<<cdiv((long)N_NODES * HID, TB), TB, 0, stream>>>(hBuf, stats, l0_g, l0_bt, xBuf);

    // ---- layers 1..2 (HID -> HID) ----
    for (int i = 0; i < 2; ++i) {
        k_prep<<<cdiv((long)N_NODES * HID, TB), TB, 0, stream>>>(yAgg, xBuf, N_NODES * HID, stats);
        k_edge<HID><<<cdiv(N_EDGES, 8), TB, 0, stream>>>(yAgg, xBuf, ei);
        k_mlp<HID><<<cdiv(N_NODES / 16, 8), TB, 0, stream>>>(
            yAgg, ws1 + (size_t)i * HID * HID, bs1 + (size_t)i * HID,
            ws2 + (size_t)i * HID * HID, bs2 + (size_t)i * HID, hBuf, stats);
        k_bn<<<cdiv((long)N_NODES * HID, TB), TB, 0, stream>>>(hBuf, stats,
                                                               gammas + (size_t)i * HID,
                                                               betas + (size_t)i * HID, xBuf);
    }

    // ---- global max pool + head ----
    k_pool_init<<<cdiv(N_GRAPHS * HID, TB), TB, 0, stream>>>(pooled);
    k_pool<<<cdiv((long)N_NODES * HID, TB), TB, 0, stream>>>(xBuf, batch, pooled);
    k_head<<<N_GRAPHS, 64, 0, stream>>>(pooled, lin1_w, lin1_b, lin2_w, lin2_b, out);
}